// ROIAlign_8581344658207
// MI455X (gfx1250) — compile-verified
//
#include <hip/hip_runtime.h>
#include <hip/hip_bf16.h>

typedef __attribute__((ext_vector_type(2))) float v2f;
typedef __attribute__((ext_vector_type(8))) float v8f;

#define T_LEN   512
#define CH      256
#define P_BINS  16
#define KDIM    128    // P_BINS * 4 samples * 2 taps
#define DSTRIDE 20     // staging row stride: 16 bins padded to 20 (16B-aligned quads)

// One block per ROI; 256 threads = 8 wave32 waves; each wave computes two
// 16-bin x 16-channel output tiles via v_wmma_f32_16x16x4_f32 accumulation
// over the K=128 (bin,sample,tap) axis. All arithmetic stays in f32.
__global__ __launch_bounds__(256)
void ROIAlign_wmma_kernel(const float* __restrict__ feat,
                          const float* __restrict__ rois,
                          float* __restrict__ out) {
    __shared__ float wlds[KDIM];                    // lerp weights (incl. 1/S, valid)
    __shared__ int   tlds[KDIM];                    // tap BYTE offsets (t*4)
    __shared__ __align__(16) float dstage[CH * DSTRIDE];

    const int n   = blockIdx.x;
    const int tid = threadIdx.x;

    const float b_f   = rois[3 * n + 0];
    const float start = rois[3 * n + 1];
    const float end   = rois[3 * n + 2];
    const int   bidx  = (int)b_f;
    const float roi_len = fmaxf(end - start, 1.0f);
    const float bin     = roi_len * (1.0f / (float)P_BINS);

    // ---- Phase 1: per-(bin,sample,tap) weight + byte offset -------------
    if (tid < KDIM) {
        const int k   = tid;
        const int p   = k >> 3;          // bin
        const int rem = k & 7;
        const int s   = rem >> 1;        // sample
        const int hi  = rem & 1;         // 0 = low tap, 1 = high tap
        const float x = start + ((float)p + ((float)s + 0.5f) * 0.25f) * bin;
        const bool valid = (x >= -1.0f) && (x <= (float)T_LEN);
        const float xc = fminf(fmaxf(x, 0.0f), (float)(T_LEN - 1));
        const int xl = (int)floorf(xc);
        const int xh = min(xl + 1, T_LEN - 1);
        const float lx = xc - (float)xl;
        float w = hi ? lx : (1.0f - lx);
        w = valid ? (w * 0.25f) : 0.0f;  // fold 1/S mean into the weight
        wlds[k] = w;
        tlds[k] = (hi ? xh : xl) << 2;   // pre-scale to bytes
    }
    __syncthreads();

    // ---- Phase 2: WMMA accumulation ------------------------------------
    const int lane = tid & 31;
    const int wave = tid >> 5;
    const int half = lane >> 4;          // 0: lanes 0-15, 1: lanes 16-31
    const int lp   = lane & 15;          // A: M row (bin) / B,D: N col (channel)

    // Uniform 64-bit base + per-lane 32-bit byte offsets -> GVS saddr+voffset
    const char* fb = (const char*)feat + (size_t)bidx * (CH * T_LEN * sizeof(float));
    const int c0 = (wave * 2 + 0) * 16 + lp;             // channel, tile 0
    const int c1 = (wave * 2 + 1) * 16 + lp;             // channel, tile 1
    const unsigned rb0 = (unsigned)c0 * (T_LEN * sizeof(float));
    const unsigned rb1 = (unsigned)c1 * (T_LEN * sizeof(float));

    v8f acc0 = {};
    v8f acc1 = {};

#pragma unroll
    for (int k0 = 0; k0 < KDIM; k0 += 4) {
        // ISA layout 16x4 f32 A / 4x16 B: VGPR0 holds K = k0 + 2*half,
        // VGPR1 holds K = k0 + 1 + 2*half; M (A) / N (B) = lane % 16.
        const int kA = k0 + 2 * half;
        const int kB = kA + 1;
        const unsigned t0 = (unsigned)tlds[kA];   // half-wave broadcast LDS reads
        const unsigned t1 = (unsigned)tlds[kB];
        const float wA = wlds[kA];
        const float wB = wlds[kB];

        // A = W^T (bins x K), block-diagonal. Both K taps of this step live in
        // bin (k0>>3) for either half-wave, so one compare feeds both selects.
        const bool own = (lp == (k0 >> 3));
        v2f a;
        a.x = own ? wA : 0.0f;
        a.y = own ? wB : 0.0f;

        // B = gathered features (K x channels), straight from L2-resident input
        v2f b0, b1;
        b0.x = *(const float*)(fb + (rb0 + t0));
        b0.y = *(const float*)(fb + (rb0 + t1));
        b1.x = *(const float*)(fb + (rb1 + t0));
        b1.y = *(const float*)(fb + (rb1 + t1));

        acc0 = __builtin_amdgcn_wmma_f32_16x16x4_f32(false, a, false, b0,
                                                     (short)0, acc0, false, false);
        acc1 = __builtin_amdgcn_wmma_f32_16x16x4_f32(false, a, false, b1,
                                                     (short)0, acc1, false, false);
    }

    // ---- Phase 3: stage D tiles in LDS (D: VGPR j -> bin j+8*half, N -> ch)
#pragma unroll
    for (int j = 0; j < 8; ++j) {
        const int p = j + 8 * half;
        dstage[c0 * DSTRIDE + p] = acc0[j];
        dstage[c1 * DSTRIDE + p] = acc1[j];
    }
    __syncthreads();

    // ---- Phase 4: coalesced b128 writeback of the (256 ch x 16 bin) slab
    float4* o4 = reinterpret_cast<float4*>(out + (size_t)n * (CH * P_BINS));
#pragma unroll
    for (int j = 0; j < 4; ++j) {
        const int g  = tid + 256 * j;    // 0..1023 quads
        const int ch = g >> 2;
        const int p4 = (g & 3) << 2;     // 16B-aligned within padded row
        o4[g] = *reinterpret_cast<const float4*>(&dstage[ch * DSTRIDE + p4]);
    }
}

extern "C" void kernel_launch(void* const* d_in, const int* in_sizes, int n_in,
                              void* d_out, int out_size, void* d_ws, size_t ws_size,
                              hipStream_t stream) {
    const float* feat = (const float*)d_in[0];   // (8, 256, 512) f32
    const float* rois = (const float*)d_in[1];   // (N, 3) f32
    // d_in[2] = feature_dim (16), d_in[3] = ratio (4): fixed by the reference
    // setup and baked in as compile-time constants (device-only scalars can't
    // be read host-side during graph capture).
    const int n_rois = in_sizes[1] / 3;          // 2048
    float* out = (float*)d_out;                  // (N, 256, 16) f32

    ROIAlign_wmma_kernel<<<n_rois, 256, 0, stream>>>(feat, rois, out);
}